// NeuralControlCritic_64055142253109
// MI455X (gfx1250) — compile-verified
//
#include <hip/hip_runtime.h>
#include <hip/hip_bf16.h>

typedef float v2f __attribute__((ext_vector_type(2)));
typedef float v8f __attribute__((ext_vector_type(8)));

#define KB   32
#define LSTR 36   // LDS row stride (floats): 16B-aligned rows, conflict-free frag reads

__device__ __forceinline__ unsigned lds_off(const void* p) {
    // generic pointers into LDS: low 32 bits are the 0-based LDS byte offset
    return (unsigned)(uintptr_t)p;
}

// C[M,N] = act( A[M,K] @ W[N,K]^T + bias[N] ), act: 0=none, 1=relu
// 256 threads = 8 waves arranged WROWS x WCOLS; wave tile (BM/WROWS) x (BN/WCOLS)
// split into 16x16 WMMA subtiles. K multiple of 32; M multiple of BM.
// Global->LDS staging uses async-to-LDS copies (ASYNCcnt) with LDS double buffering.
template<int BM, int BN, int WROWS, int WCOLS>
__global__ __launch_bounds__(256)
void gemm_wmma(const float* __restrict__ A, const float* __restrict__ W,
               const float* __restrict__ bias, float* __restrict__ C,
               int M, int N, int K, int act)
{
    constexpr int WMT = BM / WROWS;      // wave tile rows
    constexpr int WNT = BN / WCOLS;      // wave tile cols
    constexpr int MI  = WMT / 16;        // 16x16 subtiles (M)
    constexpr int NI  = WNT / 16;        // 16x16 subtiles (N)
    constexpr int NA  = (BM * KB) / (256 * 4);   // B128 copies per thread (A tile)
    constexpr int NB  = (BN * KB) / (256 * 4);   // B128 copies per thread (B tile)

    __shared__ __align__(16) float As[2][BM * LSTR];
    __shared__ __align__(16) float Bs[2][BN * LSTR];

    const int tid  = threadIdx.x;
    const int wave = tid >> 5;
    const int lane = tid & 31;
    const int wm   = wave / WCOLS;
    const int wn   = wave % WCOLS;
    const int half = lane >> 4;          // 0/1: K{0,1} vs K{2,3}
    const int l16  = lane & 15;
    const int mbase = blockIdx.y * BM;
    const int nbase = blockIdx.x * BN;

    v8f zero = {};
    v8f acc[MI][NI];
    #pragma unroll
    for (int mi = 0; mi < MI; ++mi)
        #pragma unroll
        for (int ni = 0; ni < NI; ++ni) acc[mi][ni] = zero;

    // Issue async global->LDS copies for one K-chunk into buffer p.
    auto issueLoads = [&](int k0, int p) {
        #pragma unroll
        for (int i = 0; i < NA; ++i) {
            int f = tid + i * 256;           // float4 index in BM x (KB/4)
            int r = f >> 3;                  // 8 float4 per row
            int c = (f & 7) * 4;
            const float* g = &A[(size_t)(mbase + r) * K + k0 + c];
            unsigned lo = lds_off(&As[p][r * LSTR + c]);
            asm volatile("global_load_async_to_lds_b128 %0, %1, off"
                         :: "v"(lo), "v"(g) : "memory");
        }
        #pragma unroll
        for (int i = 0; i < NB; ++i) {
            int f = tid + i * 256;
            int r = f >> 3;
            int c = (f & 7) * 4;
            int n = nbase + r;
            if (n >= N) n = N - 1;           // clamp (masked on store)
            const float* g = &W[(size_t)n * K + k0 + c];
            unsigned lo = lds_off(&Bs[p][r * LSTR + c]);
            asm volatile("global_load_async_to_lds_b128 %0, %1, off"
                         :: "v"(lo), "v"(g) : "memory");
        }
    };

    issueLoads(0, 0);
    asm volatile("s_wait_asynccnt 0x0" ::: "memory");
    __syncthreads();

    int p = 0;
    for (int k0 = 0; k0 < K; k0 += KB) {
        // prefetch next chunk into the other buffer while computing this one
        if (k0 + KB < K) issueLoads(k0 + KB, p ^ 1);

        #pragma unroll
        for (int kk = 0; kk < KB; kk += 4) {
            v2f af[MI], bf[NI];
            #pragma unroll
            for (int mi = 0; mi < MI; ++mi)
                af[mi] = *(const v2f*)&As[p][(wm * WMT + mi * 16 + l16) * LSTR + kk + half * 2];
            #pragma unroll
            for (int ni = 0; ni < NI; ++ni)
                bf[ni] = *(const v2f*)&Bs[p][(wn * WNT + ni * 16 + l16) * LSTR + kk + half * 2];
            #pragma unroll
            for (int mi = 0; mi < MI; ++mi)
                #pragma unroll
                for (int ni = 0; ni < NI; ++ni)
                    acc[mi][ni] = __builtin_amdgcn_wmma_f32_16x16x4_f32(
                        false, af[mi], false, bf[ni], (short)0, acc[mi][ni], false, false);
        }

        // my async copies into buffer p^1 are done; barrier covers all waves
        asm volatile("s_wait_asynccnt 0x0" ::: "memory");
        __syncthreads();
        p ^= 1;
    }

    // store: VGPR v -> M = v + 8*half within subtile, N = l16
    #pragma unroll
    for (int ni = 0; ni < NI; ++ni) {
        int n = nbase + wn * WNT + ni * 16 + l16;
        if (n < N) {
            float bv = bias ? bias[n] : 0.0f;
            #pragma unroll
            for (int mi = 0; mi < MI; ++mi) {
                #pragma unroll
                for (int v = 0; v < 8; ++v) {
                    int m = mbase + wm * WMT + mi * 16 + half * 8 + v;
                    if (m < M) {
                        float val = acc[mi][ni][v] + bv;
                        if (act == 1) val = fmaxf(val, 0.0f);
                        C[(size_t)m * N + n] = val;
                    }
                }
            }
        }
    }
}

// Mamba2 middle: depthwise conv(4) + SiLU, softplus(dt), 32-step SSM scan with
// register-resident state, D-skip, SiLU(z) gating, RMSNorm * norm_w.
// One block per batch element, 1024 threads.
// zx: (B*32, 772)  [z:0..255 | xBC:256..767 | dt:768..771] ;  yN: (B*32, 256)
__global__ __launch_bounds__(1024)
void mamba_mid(const float* __restrict__ zx,
               const float* __restrict__ conv_w,   // (512,4)
               const float* __restrict__ conv_b,   // (512)
               const float* __restrict__ dt_bias,  // (4)
               const float* __restrict__ A_log,    // (4)
               const float* __restrict__ Dv,       // (4)
               const float* __restrict__ norm_w,   // (256)
               float* __restrict__ yN)
{
    __shared__ float sXBC[32][512];
    __shared__ float sY[32][256];
    __shared__ float sdt[32][4];
    __shared__ float sdA[32][4];

    const int tid = threadIdx.x;
    const size_t rowbase = (size_t)blockIdx.x * 32;

    // Phase 1: causal depthwise conv + bias + SiLU over 32x512 channels.
    #pragma unroll
    for (int i = 0; i < 16; ++i) {
        int e = tid + i * 1024;
        int t = e >> 9;
        int c = e & 511;
        float a = conv_b[c];
        #pragma unroll
        for (int j = 0; j < 4; ++j) {
            int tt = t - 3 + j;
            if (tt >= 0) a += conv_w[c * 4 + j] * zx[(rowbase + tt) * 772 + 256 + c];
        }
        sXBC[t][c] = a / (1.0f + expf(-a));
    }
    if (tid < 128) {
        int t = tid >> 2, hh = tid & 3;
        float dtr = zx[(rowbase + t) * 772 + 768 + hh] + dt_bias[hh];
        float dts = (dtr > 20.0f) ? dtr : log1pf(expf(dtr));
        sdt[t][hh] = dts;
        sdA[t][hh] = expf(-expf(A_log[hh]) * dts);
    }
    __syncthreads();

    // Phase 2: scan. Thread owns (head,p) = tid>>2 and 32 of 128 state cols.
    const int hp   = tid >> 2;
    const int head = hp >> 6;
    const int nq   = tid & 3;
    const int nb   = nq * 32;
    const float Dh = Dv[head];

    float st[32];
    #pragma unroll
    for (int j = 0; j < 32; ++j) st[j] = 0.0f;

    for (int t = 0; t < 32; ++t) {
        float xv   = sXBC[t][hp];
        float dav  = sdA[t][head];
        float coef = sdt[t][head] * xv;
        float a = 0.0f;
        #pragma unroll
        for (int j = 0; j < 32; ++j) {
            float Bn = sXBC[t][256 + nb + j];
            float Cn = sXBC[t][384 + nb + j];
            st[j] = st[j] * dav + coef * Bn;
            a += st[j] * Cn;
        }
        a += __shfl_xor(a, 1);
        a += __shfl_xor(a, 2);
        if (nq == 0) sY[t][hp] = a + Dh * xv;
    }
    __syncthreads();

    // Phase 3: gating + RMSNorm, one wave per timestep.
    const int w    = tid >> 5;
    const int lane = tid & 31;
    float g[8];
    float ss = 0.0f;
    #pragma unroll
    for (int i = 0; i < 8; ++i) {
        int c = lane + i * 32;
        float z = zx[(rowbase + w) * 772 + c];
        float y = sY[w][c] * (z / (1.0f + expf(-z)));
        g[i] = y;
        ss += y * y;
    }
    #pragma unroll
    for (int off = 16; off > 0; off >>= 1) ss += __shfl_xor(ss, off);
    float rs = rsqrtf(ss * (1.0f / 256.0f) + 1e-5f);
    #pragma unroll
    for (int i = 0; i < 8; ++i) {
        int c = lane + i * 32;
        yN[(rowbase + w) * 256 + c] = g[i] * rs * norm_w[c];
    }
}

// M=256 layers: 64x128 block tile (2x4 waves, 32x32 wave tile) -> more blocks.
static inline void gemm_small(const float* A, const float* W, const float* bias,
                              float* C, int M, int N, int K, int act, hipStream_t s)
{
    dim3 grid((N + 127) / 128, (M + 63) / 64);
    gemm_wmma<64, 128, 2, 4><<<grid, 256, 0, s>>>(A, W, bias, C, M, N, K, act);
}

// M=8192 layers: 128x128 block tile (4x2 waves, 32x64 wave tile).
static inline void gemm_big(const float* A, const float* W, const float* bias,
                            float* C, int M, int N, int K, int act, hipStream_t s)
{
    dim3 grid((N + 127) / 128, (M + 127) / 128);
    gemm_wmma<128, 128, 4, 2><<<grid, 256, 0, s>>>(A, W, bias, C, M, N, K, act);
}

extern "C" void kernel_launch(void* const* d_in, const int* in_sizes, int n_in,
                              void* d_out, int out_size, void* d_ws, size_t ws_size,
                              hipStream_t stream)
{
    const float* x          = (const float*)d_in[0];   // (256,4096)
    const float* mlp_in_W   = (const float*)d_in[1];   // (2,3,4096,4096)
    const float* mlp_in_b   = (const float*)d_in[2];   // (2,3,4096)
    const float* mlp_out_W1 = (const float*)d_in[3];   // (2,4096,4096)
    const float* mlp_out_b1 = (const float*)d_in[4];   // (2,4096)
    const float* mlp_out_W2 = (const float*)d_in[5];   // (2,410,4096)
    const float* mlp_out_b2 = (const float*)d_in[6];   // (2,410)
    const float* m_in_w     = (const float*)d_in[7];   // (2,2,772,128)
    const float* m_conv_w   = (const float*)d_in[8];   // (2,2,512,4)
    const float* m_conv_b   = (const float*)d_in[9];   // (2,2,512)
    const float* m_dt_bias  = (const float*)d_in[10];  // (2,2,4)
    const float* m_A_log    = (const float*)d_in[11];  // (2,2,4)
    const float* m_D        = (const float*)d_in[12];  // (2,2,4)
    const float* m_norm_w   = (const float*)d_in[13];  // (2,2,256)
    const float* m_out_w    = (const float*)d_in[14];  // (2,2,128,256)
    float* out = (float*)d_out;
    float* ws  = (float*)d_ws;

    const int Bsz = 256, HID = 4096, SEQROWS = 256 * 32, DPROJ = 772, OUT_DIM = 410;

    float* H0 = ws;                              // 256*4096
    float* H1 = H0 + (size_t)Bsz * HID;          // 256*4096
    float* ZX = H1 + (size_t)Bsz * HID;          // 8192*772
    float* YN = ZX + (size_t)SEQROWS * DPROJ;    // 8192*256   (~42 MB total)

    for (int i = 0; i < 2; ++i) {
        // --- 3 input MLP layers (4096x4096) ---
        gemm_small(x,  mlp_in_W + ((size_t)i * 3 + 0) * HID * HID,
                   mlp_in_b + ((size_t)i * 3 + 0) * HID, H0, Bsz, HID, HID, 1, stream);
        gemm_small(H0, mlp_in_W + ((size_t)i * 3 + 1) * HID * HID,
                   mlp_in_b + ((size_t)i * 3 + 1) * HID, H1, Bsz, HID, HID, 1, stream);
        gemm_small(H1, mlp_in_W + ((size_t)i * 3 + 2) * HID * HID,
                   mlp_in_b + ((size_t)i * 3 + 2) * HID, H0, Bsz, HID, HID, 0, stream);

        // --- 2 Mamba2 layers; H* reinterpreted as (8192,128) ---
        float* cur = H0;
        float* oth = H1;
        for (int layer = 0; layer < 2; ++layer) {
            size_t li = (size_t)i * 2 + layer;
            gemm_big(cur, m_in_w + li * DPROJ * 128, nullptr, ZX,
                     SEQROWS, DPROJ, 128, 0, stream);
            mamba_mid<<<256, 1024, 0, stream>>>(ZX,
                m_conv_w + li * 512 * 4, m_conv_b + li * 512,
                m_dt_bias + li * 4, m_A_log + li * 4, m_D + li * 4,
                m_norm_w + li * 256, YN);
            gemm_big(YN, m_out_w + li * 128 * 256, nullptr, oth,
                     SEQROWS, 128, 256, 0, stream);
            float* tmp = cur; cur = oth; oth = tmp;
        }

        // --- output MLP ---
        gemm_small(cur, mlp_out_W1 + (size_t)i * HID * HID,
                   mlp_out_b1 + (size_t)i * HID, oth, Bsz, HID, HID, 1, stream);
        gemm_small(oth, mlp_out_W2 + (size_t)i * OUT_DIM * HID,
                   mlp_out_b2 + (size_t)i * OUT_DIM,
                   out + (size_t)i * Bsz * OUT_DIM, Bsz, OUT_DIM, HID, 0, stream);
    }
}